// NiftyregImageResamplingLayer_13606456393817
// MI455X (gfx1250) — compile-verified
//
#include <hip/hip_runtime.h>
#include <math.h>

// Trilinear resampling (LINEAR, ZERO boundary) for MI455X / gfx1250.
//
// inputs:      [B=2, S=144, S, S, C=2] f32   (47.8 MB -> resident in 192MB L2)
// deformation: [B, S, S, S, 3] f32           (streamed once, NT)
// out:         [B, S, S, S, C] f32           (written once, NT)
//
// Memory-bound gather kernel: ~84 B traffic vs ~30 FLOPs per sample point
// (AI ~0.35 FLOP/B) -> bandwidth/roofline bound, HBM floor ~7us at 23.3 TB/s.
// Strategy: async global->LDS staging of the AoS deformation stream (coalesced,
// no VGPR round-trip, ASYNCcnt-tracked), RT-cached b64 corner gathers that hit
// the L2-resident volume, NT store of the write-once output.

typedef float v2f __attribute__((ext_vector_type(2)));
typedef __attribute__((address_space(1))) int* gint_p;  // global (AS1)
typedef __attribute__((address_space(3))) int* lint_p;  // LDS (AS3)

#define S_DIM 144
#define C_DIM 2
#define BLOCK 256

#if defined(__gfx1250__) && __has_builtin(__builtin_amdgcn_global_load_async_to_lds_b32)
#define HAVE_ASYNC_LDS 1
#else
#define HAVE_ASYNC_LDS 0
#endif

// cpol: TH bits [2:0]; 1 = non-temporal (read-once deformation stream).
#define CPOL_NT 1

__global__ __launch_bounds__(BLOCK) void trilinear_zero_kernel(
    const float* __restrict__ inputs,
    const float* __restrict__ def,
    float* __restrict__ out,
    int n_points)  // B * S^3
{
  __shared__ float sdef[BLOCK * 3];

  const int t          = threadIdx.x;
  const int blockStart = blockIdx.x * BLOCK;           // first sample point of block
  const int gbase      = blockStart * 3;               // first def float of block
  const int total_def  = n_points * 3;                 // ~17.9M, fits int32

  // ---- Stage this block's deformation slice (768 floats) into LDS, coalesced.
#if HAVE_ASYNC_LDS
  if (gbase + 3 * BLOCK <= total_def) {
    // Full block (common case): uniform path, EXEC stays all-ones, the three
    // async copies issue back-to-back on ASYNCcnt.
#pragma unroll
    for (int k = 0; k < 3; ++k) {
      __builtin_amdgcn_global_load_async_to_lds_b32(
          (gint_p)(def + gbase + t + k * BLOCK),
          (lint_p)(&sdef[t + k * BLOCK]),
          /*offset=*/0, /*cpol=*/CPOL_NT);
    }
  } else {
    // Tail block: per-lane guard (exec-masked copies).
#pragma unroll
    for (int k = 0; k < 3; ++k) {
      int g = gbase + t + k * BLOCK;
      if (g < total_def) {
        __builtin_amdgcn_global_load_async_to_lds_b32(
            (gint_p)(def + g),
            (lint_p)(&sdef[t + k * BLOCK]),
            /*offset=*/0, /*cpol=*/CPOL_NT);
      }
    }
  }
#if __has_builtin(__builtin_amdgcn_s_wait_asynccnt)
  __builtin_amdgcn_s_wait_asynccnt(0);
#else
  asm volatile("s_wait_asynccnt 0" ::: "memory");
#endif
#else
  // Fallback: coalesced non-temporal loads + LDS stores.
#pragma unroll
  for (int k = 0; k < 3; ++k) {
    int g = gbase + t + k * BLOCK;
    if (g < total_def) sdef[t + k * BLOCK] = __builtin_nontemporal_load(def + g);
  }
#endif
  __syncthreads();

  const int idx = blockStart + t;
  if (idx >= n_points) return;

  // Per-lane coords from LDS (stride-3 over 64 banks: conflict-free).
  const float cx = sdef[3 * t + 0];
  const float cy = sdef[3 * t + 1];
  const float cz = sdef[3 * t + 2];

  const int   vol = S_DIM * S_DIM * S_DIM;
  const int   b   = idx / vol;  // batch index (const-divide -> mul/shift)

  const float fbx = floorf(cx), fby = floorf(cy), fbz = floorf(cz);
  const int   x0 = (int)fbx, y0 = (int)fby, z0 = (int)fbz;
  const int   x1 = x0 + 1,  y1 = y0 + 1,  z1 = z0 + 1;
  const float fx = cx - fbx, fy = cy - fby, fz = cz - fbz;

  const bool vx0 = (unsigned)x0 < (unsigned)S_DIM;
  const bool vx1 = (unsigned)x1 < (unsigned)S_DIM;
  const bool vy0 = (unsigned)y0 < (unsigned)S_DIM;
  const bool vy1 = (unsigned)y1 < (unsigned)S_DIM;
  const bool vz0 = (unsigned)z0 < (unsigned)S_DIM;
  const bool vz1 = (unsigned)z1 < (unsigned)S_DIM;

  // Clamped indices for address math (loads are predicated; clamp keeps
  // pointer arithmetic in-range, matching the reference's clip()).
  const int cx0 = min(max(x0, 0), S_DIM - 1), cx1 = min(max(x1, 0), S_DIM - 1);
  const int cy0 = min(max(y0, 0), S_DIM - 1), cy1 = min(max(y1, 0), S_DIM - 1);
  const int cz0 = min(max(z0, 0), S_DIM - 1), cz1 = min(max(z1, 0), S_DIM - 1);

  const int sZ = C_DIM;                  // 2
  const int sY = S_DIM * C_DIM;          // 288
  const int sX = S_DIM * S_DIM * C_DIM;  // 41472
  const float* bp = inputs + (size_t)b * (size_t)vol * C_DIM;

  const int oX0 = cx0 * sX, oX1 = cx1 * sX;
  const int oY0 = cy0 * sY, oY1 = cy1 * sY;
  const int oZ0 = cz0 * sZ, oZ1 = cz1 * sZ;

  // 8 predicated b64 gathers (both channels per corner; offsets are even ->
  // 8-byte aligned). These default to RT caching so the volume stays in L2.
  v2f v000 = {0.f, 0.f}, v001 = {0.f, 0.f}, v010 = {0.f, 0.f}, v011 = {0.f, 0.f};
  v2f v100 = {0.f, 0.f}, v101 = {0.f, 0.f}, v110 = {0.f, 0.f}, v111 = {0.f, 0.f};
  if (vx0 & vy0 & vz0) v000 = *(const v2f*)(bp + oX0 + oY0 + oZ0);
  if (vx0 & vy0 & vz1) v001 = *(const v2f*)(bp + oX0 + oY0 + oZ1);
  if (vx0 & vy1 & vz0) v010 = *(const v2f*)(bp + oX0 + oY1 + oZ0);
  if (vx0 & vy1 & vz1) v011 = *(const v2f*)(bp + oX0 + oY1 + oZ1);
  if (vx1 & vy0 & vz0) v100 = *(const v2f*)(bp + oX1 + oY0 + oZ0);
  if (vx1 & vy0 & vz1) v101 = *(const v2f*)(bp + oX1 + oY0 + oZ1);
  if (vx1 & vy1 & vz0) v110 = *(const v2f*)(bp + oX1 + oY1 + oZ0);
  if (vx1 & vy1 & vz1) v111 = *(const v2f*)(bp + oX1 + oY1 + oZ1);

  const float wx0 = 1.f - fx, wy0 = 1.f - fy, wz0 = 1.f - fz;
  const float w000 = wx0 * wy0 * wz0;
  const float w001 = wx0 * wy0 * fz;
  const float w010 = wx0 * fy  * wz0;
  const float w011 = wx0 * fy  * fz;
  const float w100 = fx  * wy0 * wz0;
  const float w101 = fx  * wy0 * fz;
  const float w110 = fx  * fy  * wz0;
  const float w111 = fx  * fy  * fz;

  v2f acc;
  acc.x = w000 * v000.x + w001 * v001.x + w010 * v010.x + w011 * v011.x +
          w100 * v100.x + w101 * v101.x + w110 * v110.x + w111 * v111.x;
  acc.y = w000 * v000.y + w001 * v001.y + w010 * v010.y + w011 * v011.y +
          w100 * v100.y + w101 * v101.y + w110 * v110.y + w111 * v111.y;

  // Write-once output: non-temporal so it doesn't evict the L2-resident volume.
  __builtin_nontemporal_store(acc, (v2f*)out + idx);
}

extern "C" void kernel_launch(void* const* d_in, const int* in_sizes, int n_in,
                              void* d_out, int out_size, void* d_ws, size_t ws_size,
                              hipStream_t stream) {
  const float* inputs = (const float*)d_in[0];  // [B,S,S,S,C] f32
  const float* def    = (const float*)d_in[1];  // [B,S,S,S,3] f32
  float* out          = (float*)d_out;          // [B,S,S,S,C] f32

  const int n_points = in_sizes[1] / 3;  // B * S^3 = 5,971,968
  const int blocks   = (n_points + BLOCK - 1) / BLOCK;

  trilinear_zero_kernel<<<blocks, BLOCK, 0, stream>>>(inputs, def, out, n_points);
}